// pointnet2_cls_ssg_2181843386699
// MI455X (gfx1250) — compile-verified
//
#include <hip/hip_runtime.h>
#include <hip/hip_bf16.h>

typedef __attribute__((ext_vector_type(2))) float v2f;
typedef __attribute__((ext_vector_type(8))) float v8f;

// ---------------------------------------------------------------------------
// Farthest point sampling: one block per batch, one thread per point.
// blockDim.x == N (power of two: 1024 or 512). Emits `far` before updating,
// matching the reference scan semantics (idx[0] == 0).
// ---------------------------------------------------------------------------
#define FPS_MAX_N 1024
__global__ __launch_bounds__(1024)
void pn2_fps_kernel(const float* __restrict__ xyz, int* __restrict__ out_idx,
                    int N, int M) {
  __shared__ float sx[FPS_MAX_N], sy[FPS_MAX_N], sz[FPS_MAX_N];
  __shared__ float rv[FPS_MAX_N];
  __shared__ int   ri[FPS_MAX_N];
  const int b = blockIdx.x;
  const int t = threadIdx.x;
  const float* base = xyz + (size_t)b * N * 3;
  float px = base[t * 3 + 0];
  float py = base[t * 3 + 1];
  float pz = base[t * 3 + 2];
  sx[t] = px; sy[t] = py; sz[t] = pz;
  float dist = 1e10f;
  int far = 0;
  __syncthreads();
  for (int it = 0; it < M; ++it) {
    if (t == 0) out_idx[b * M + it] = far;
    const float fx = sx[far], fy = sy[far], fz = sz[far];
    const float dx = px - fx, dy = py - fy, dz = pz - fz;
    const float d = dx * dx + dy * dy + dz * dz;
    dist = fminf(dist, d);
    rv[t] = dist; ri[t] = t;
    __syncthreads();
    for (int s = N >> 1; s > 0; s >>= 1) {
      if (t < s) {
        const float v2 = rv[t + s]; const int i2 = ri[t + s];
        if (v2 > rv[t] || (v2 == rv[t] && i2 < ri[t])) { rv[t] = v2; ri[t] = i2; }
      }
      __syncthreads();
    }
    far = ri[0];
    __syncthreads();
  }
}

// Gather selected centroids: out[b,m,:] = xyz[b, idx[b,m], :]
__global__ void pn2_gather_xyz_kernel(const float* __restrict__ xyz,
                                      const int* __restrict__ idx,
                                      float* __restrict__ out,
                                      int N, int M, int total) {
  const int i = blockIdx.x * blockDim.x + threadIdx.x;
  if (i >= total) return;
  const int b = i / M;
  const int j = idx[i];
  const float* p = xyz + ((size_t)b * N + j) * 3;
  out[i * 3 + 0] = p[0];
  out[i * 3 + 1] = p[1];
  out[i * 3 + 2] = p[2];
}

// Ball query: first K indices (in increasing order) with d2 <= r2; pad with first.
__global__ void pn2_ballq_kernel(const float* __restrict__ xyz,
                                 const float* __restrict__ newxyz,
                                 int* __restrict__ gidx,
                                 int N, int M, int K, float r2, int total) {
  const int i = blockIdx.x * blockDim.x + threadIdx.x;
  if (i >= total) return;
  const int b = i / M;
  const float* base = xyz + (size_t)b * N * 3;
  const float cx = newxyz[i * 3 + 0];
  const float cy = newxyz[i * 3 + 1];
  const float cz = newxyz[i * 3 + 2];
  int* out = gidx + (size_t)i * K;
  int cnt = 0, first = 0;
  for (int j = 0; j < N && cnt < K; ++j) {
    const float dx = base[j * 3 + 0] - cx;
    const float dy = base[j * 3 + 1] - cy;
    const float dz = base[j * 3 + 2] - cz;
    const float d2 = dx * dx + dy * dy + dz * dz;
    if (d2 <= r2) { if (cnt == 0) first = j; out[cnt++] = j; }
  }
  for (int k = cnt; k < K; ++k) out[k] = first;
}

// SA1 grouping (padded to 8 channels): X[i,0:3]=xyz[j]-center, X[i,3:6]=points[j], X[i,6:8]=0
__global__ void pn2_group1_kernel(const float* __restrict__ xyz,
                                  const float* __restrict__ points,
                                  const float* __restrict__ newxyz,
                                  const int* __restrict__ gidx,
                                  float* __restrict__ X,
                                  int N, int M, int K, int total) {
  const int i = blockIdx.x * blockDim.x + threadIdx.x;
  if (i >= total) return;
  const int j = gidx[i];
  const int bm = i / K;
  const int b = bm / M;
  const float* p = xyz + ((size_t)b * N + j) * 3;
  const float* q = points + ((size_t)b * N + j) * 3;
  float* row = X + (size_t)i * 8;
  row[0] = p[0] - newxyz[bm * 3 + 0];
  row[1] = p[1] - newxyz[bm * 3 + 1];
  row[2] = p[2] - newxyz[bm * 3 + 2];
  row[3] = q[0]; row[4] = q[1]; row[5] = q[2];
  row[6] = 0.0f; row[7] = 0.0f;
}

// SA2 grouping (padded to 132): X[i,0:3]=nx1[j]-center, X[i,3:131]=pooled1[b,j,:], X[i,131]=0
__global__ void pn2_group2_kernel(const float* __restrict__ nx1,
                                  const float* __restrict__ pooled1,
                                  const float* __restrict__ newxyz,
                                  const int* __restrict__ gidx,
                                  float* __restrict__ X,
                                  int N, int M, int K, int C, int total) {
  const int i = blockIdx.x * blockDim.x + threadIdx.x;
  if (i >= total) return;
  const int j = gidx[i];
  const int bm = i / K;
  const int b = bm / M;
  const float* p = nx1 + ((size_t)b * N + j) * 3;
  const float* f = pooled1 + ((size_t)b * N + j) * C;
  float* row = X + (size_t)i * (4 + C);      // 132
  row[0] = p[0] - newxyz[bm * 3 + 0];
  row[1] = p[1] - newxyz[bm * 3 + 1];
  row[2] = p[2] - newxyz[bm * 3 + 2];
  for (int c = 0; c < C; ++c) row[3 + c] = f[c];
  row[3 + C] = 0.0f;
}

// SA3 group-all (padded to 260): X[i,0:3]=nx2[i], X[i,3:259]=pooled2[i,:], X[i,259]=0
__global__ void pn2_group3_kernel(const float* __restrict__ nx2,
                                  const float* __restrict__ pooled2,
                                  float* __restrict__ X, int C, int total) {
  const int i = blockIdx.x * blockDim.x + threadIdx.x;
  if (i >= total) return;
  float* row = X + (size_t)i * (4 + C);      // 260
  row[0] = nx2[i * 3 + 0];
  row[1] = nx2[i * 3 + 1];
  row[2] = nx2[i * 3 + 2];
  const float* f = pooled2 + (size_t)i * C;
  for (int c = 0; c < C; ++c) row[3 + c] = f[c];
  row[3 + C] = 0.0f;
}

// Transpose + zero-pad a weight matrix: Wt[(Cout,Cin4)] with Wt[n][k] = W[k][n],
// zero for k >= Cin. Makes the per-lane WMMA B pair (k, k+1) contiguous in memory.
__global__ void pn2_padwt_kernel(const float* __restrict__ W, float* __restrict__ Wt,
                                 int Cin, int Cin4, int Cout) {
  const int i = blockIdx.x * blockDim.x + threadIdx.x;
  const int total = Cout * Cin4;
  if (i >= total) return;
  const int n = i / Cin4, k = i % Cin4;
  Wt[i] = (k < Cin) ? W[(size_t)k * Cout + n] : 0.0f;
}

// ---------------------------------------------------------------------------
// fp32 WMMA GEMM: Y[R,Cout] = X[R,Cin] @ W[Cin,Cout], with W pre-transposed as
// Wt[Cout][Cin] (Cin % 4 == 0, Cin >= 8 after padding). Block = (32,4): 4
// waves, each wave computes a 16x64 strip (4 WMMA tiles sharing one A pair per
// K-step). Explicit 2-stage software pipeline: issue next K-step's 5 b64 loads
// before the current step's 4 WMMAs so load latency overlaps the matrix pipe.
// Cout % 64 == 0; R % 16 == 0.
// ---------------------------------------------------------------------------
__global__ __launch_bounds__(128)
void pn2_wmma_gemm_kernel(const float* __restrict__ X,
                          const float* __restrict__ Wt,
                          float* __restrict__ Y,
                          int R, int Cin, int Cout) {
  const int lane = threadIdx.x;            // 0..31
  const int half = lane >> 4;              // 0 -> K+{0,1}, 1 -> K+{2,3}
  const int lm = lane & 15;
  const int mt = blockIdx.x * 4 + threadIdx.y;   // 16-row tile index
  if (mt * 16 >= R) return;                // wave-uniform: EXEC stays all-ones
  const int n0 = blockIdx.y * 64;
  const float* xrow = X + (size_t)(mt * 16 + lm) * Cin + 2 * half;
  const float* wr0 = Wt + (size_t)(n0 + lm) * Cin + 2 * half;
  const float* wr1 = wr0 + (size_t)16 * Cin;
  const float* wr2 = wr0 + (size_t)32 * Cin;
  const float* wr3 = wr0 + (size_t)48 * Cin;
  v8f acc0 = {}, acc1 = {}, acc2 = {}, acc3 = {};
  // pipeline prologue: K-step 0 operands
  v2f a  = *(const v2f*)(xrow);
  v2f b0 = *(const v2f*)(wr0);
  v2f b1 = *(const v2f*)(wr1);
  v2f b2 = *(const v2f*)(wr2);
  v2f b3 = *(const v2f*)(wr3);
  for (int k0 = 4; k0 < Cin; k0 += 4) {
    // issue next step's loads first, then consume current registers
    const v2f an  = *(const v2f*)(xrow + k0);
    const v2f bn0 = *(const v2f*)(wr0 + k0);
    const v2f bn1 = *(const v2f*)(wr1 + k0);
    const v2f bn2 = *(const v2f*)(wr2 + k0);
    const v2f bn3 = *(const v2f*)(wr3 + k0);
    acc0 = __builtin_amdgcn_wmma_f32_16x16x4_f32(false, a, false, b0, (short)0, acc0, false, false);
    acc1 = __builtin_amdgcn_wmma_f32_16x16x4_f32(false, a, false, b1, (short)0, acc1, false, false);
    acc2 = __builtin_amdgcn_wmma_f32_16x16x4_f32(false, a, false, b2, (short)0, acc2, false, false);
    acc3 = __builtin_amdgcn_wmma_f32_16x16x4_f32(false, a, false, b3, (short)0, acc3, false, false);
    a = an; b0 = bn0; b1 = bn1; b2 = bn2; b3 = bn3;
  }
  // pipeline epilogue
  acc0 = __builtin_amdgcn_wmma_f32_16x16x4_f32(false, a, false, b0, (short)0, acc0, false, false);
  acc1 = __builtin_amdgcn_wmma_f32_16x16x4_f32(false, a, false, b1, (short)0, acc1, false, false);
  acc2 = __builtin_amdgcn_wmma_f32_16x16x4_f32(false, a, false, b2, (short)0, acc2, false, false);
  acc3 = __builtin_amdgcn_wmma_f32_16x16x4_f32(false, a, false, b3, (short)0, acc3, false, false);
  // C/D layout: VGPR v, lanes 0-15 -> M=v, lanes 16-31 -> M=v+8; N = lane&15.
  float* yb = Y + (size_t)(mt * 16 + 8 * half) * Cout + n0 + lm;
#pragma unroll
  for (int v = 0; v < 8; ++v) {
    float* yr = yb + (size_t)v * Cout;
    yr[0]  = acc0[v];
    yr[16] = acc1[v];
    yr[32] = acc2[v];
    yr[48] = acc3[v];
  }
}

// Per-channel batchnorm stats over all R rows: one block per channel.
__global__ __launch_bounds__(256)
void pn2_bn_stats_kernel(const float* __restrict__ Y, int R, int C,
                         float* __restrict__ mean, float* __restrict__ rinv) {
  __shared__ float ss[256], sq[256];
  const int c = blockIdx.x;
  const int t = threadIdx.x;
  float s = 0.0f, q = 0.0f;
  for (int r = t; r < R; r += 256) {
    const float v = Y[(size_t)r * C + c];
    s += v; q += v * v;
  }
  ss[t] = s; sq[t] = q;
  __syncthreads();
  for (int w = 128; w > 0; w >>= 1) {
    if (t < w) { ss[t] += ss[t + w]; sq[t] += sq[t + w]; }
    __syncthreads();
  }
  if (t == 0) {
    const float m = ss[0] / (float)R;
    const float var = sq[0] / (float)R - m * m;
    mean[c] = m;
    rinv[c] = rsqrtf(var + 1e-5f);
  }
}

// In-place normalize + scale/shift + ReLU.
__global__ void pn2_bn_relu_kernel(float* __restrict__ Y, int C,
                                   const float* __restrict__ mean,
                                   const float* __restrict__ rinv,
                                   const float* __restrict__ g,
                                   const float* __restrict__ b,
                                   long long total) {
  const long long i = (long long)blockIdx.x * blockDim.x + threadIdx.x;
  if (i >= total) return;
  const int c = (int)(i % C);
  const float v = (Y[i] - mean[c]) * rinv[c] * g[c] + b[c];
  Y[i] = fmaxf(v, 0.0f);
}

// Max-pool over the K (group) axis: (BM, K, C) -> (BM, C)
__global__ void pn2_maxpool_kernel(const float* __restrict__ Y,
                                   float* __restrict__ out,
                                   int K, int C, long long total) {
  const long long i = (long long)blockIdx.x * blockDim.x + threadIdx.x;
  if (i >= total) return;
  const long long bm = i / C;
  const int c = (int)(i % C);
  const float* p = Y + (size_t)bm * K * C + c;
  float mx = p[0];
  for (int k = 1; k < K; ++k) mx = fmaxf(mx, p[(size_t)k * C]);
  out[i] = mx;
}

// Final classifier: out[32,40] = h[32,256] @ W[256,40] + bias
__global__ void pn2_cls_kernel(const float* __restrict__ h,
                               const float* __restrict__ W,
                               const float* __restrict__ bias,
                               float* __restrict__ out) {
  const int i = blockIdx.x * blockDim.x + threadIdx.x;
  if (i >= 32 * 40) return;
  const int r = i / 40, c = i % 40;
  float s = bias[c];
  for (int k = 0; k < 256; ++k) s += h[r * 256 + k] * W[k * 40 + c];
  out[i] = s;
}

extern "C" void kernel_launch(void* const* d_in, const int* in_sizes, int n_in,
                              void* d_out, int out_size, void* d_ws, size_t ws_size,
                              hipStream_t stream) {
  (void)in_sizes; (void)n_in; (void)out_size; (void)ws_size;
  const float* xyz    = (const float*)d_in[0];   // (32,1024,3)
  const float* points = (const float*)d_in[1];   // (32,1024,3)
  const float* w11 = (const float*)d_in[2];  const float* g11 = (const float*)d_in[3];  const float* b11 = (const float*)d_in[4];
  const float* w12 = (const float*)d_in[5];  const float* g12 = (const float*)d_in[6];  const float* b12 = (const float*)d_in[7];
  const float* w13 = (const float*)d_in[8];  const float* g13 = (const float*)d_in[9];  const float* b13 = (const float*)d_in[10];
  const float* w21 = (const float*)d_in[11]; const float* g21 = (const float*)d_in[12]; const float* b21 = (const float*)d_in[13];
  const float* w22 = (const float*)d_in[14]; const float* g22 = (const float*)d_in[15]; const float* b22 = (const float*)d_in[16];
  const float* w23 = (const float*)d_in[17]; const float* g23 = (const float*)d_in[18]; const float* b23 = (const float*)d_in[19];
  const float* w31 = (const float*)d_in[20]; const float* g31 = (const float*)d_in[21]; const float* b31 = (const float*)d_in[22];
  const float* w32 = (const float*)d_in[23]; const float* g32 = (const float*)d_in[24]; const float* b32 = (const float*)d_in[25];
  const float* w33 = (const float*)d_in[26]; const float* g33 = (const float*)d_in[27]; const float* b33 = (const float*)d_in[28];
  const float* fc1 = (const float*)d_in[29]; const float* gh1 = (const float*)d_in[30]; const float* bh1 = (const float*)d_in[31];
  const float* fc2 = (const float*)d_in[32]; const float* gh2 = (const float*)d_in[33]; const float* bh2 = (const float*)d_in[34];
  const float* clw = (const float*)d_in[35]; const float* clb = (const float*)d_in[36];
  float* out = (float*)d_out;

  const int B = 32, N = 1024;
  const int M1 = 512, K1 = 32, R1 = B * M1 * K1;  // 524288
  const int M2 = 128, K2 = 64, R2 = B * M2 * K2;  // 262144
  const int R3 = B * 128;                          // 4096

  // ---- workspace layout ----
  float* P0 = (float*)d_ws;                // 524288*128 floats (268 MB)
  float* P1 = P0 + (size_t)67108864;       // same size
  char* sm = (char*)(P1 + (size_t)67108864);
  int*   fps1    = (int*)sm;     sm += (size_t)B * M1 * 4;
  float* nx1     = (float*)sm;   sm += (size_t)B * M1 * 3 * 4;
  int*   gidx1   = (int*)sm;     sm += (size_t)B * M1 * K1 * 4;
  float* pooled1 = (float*)sm;   sm += (size_t)B * M1 * 128 * 4;
  int*   fps2    = (int*)sm;     sm += (size_t)B * M2 * 4;
  float* nx2     = (float*)sm;   sm += (size_t)B * M2 * 3 * 4;
  int*   gidx2   = (int*)sm;     sm += (size_t)B * M2 * K2 * 4;
  float* pooled2 = (float*)sm;   sm += (size_t)B * M2 * 256 * 4;
  float* pooled3 = (float*)sm;   sm += (size_t)B * 1024 * 4;
  float* h1      = (float*)sm;   sm += (size_t)B * 512 * 4;
  float* h2      = (float*)sm;   sm += (size_t)B * 256 * 4;
  float* meanb   = (float*)sm;   sm += 1024 * 4;
  float* rinvb   = (float*)sm;   sm += 1024 * 4;
  // transposed (+K-padded) weights: Wt[Cout][Cin4]
  float* wt11 = (float*)sm;  sm += (size_t)64   * 8    * 4;
  float* wt12 = (float*)sm;  sm += (size_t)64   * 64   * 4;
  float* wt13 = (float*)sm;  sm += (size_t)128  * 64   * 4;
  float* wt21 = (float*)sm;  sm += (size_t)128  * 132  * 4;
  float* wt22 = (float*)sm;  sm += (size_t)128  * 128  * 4;
  float* wt23 = (float*)sm;  sm += (size_t)256  * 128  * 4;
  float* wt31 = (float*)sm;  sm += (size_t)256  * 260  * 4;
  float* wt32 = (float*)sm;  sm += (size_t)512  * 256  * 4;
  float* wt33 = (float*)sm;  sm += (size_t)1024 * 512  * 4;
  float* wtf1 = (float*)sm;  sm += (size_t)512  * 1024 * 4;
  float* wtf2 = (float*)sm;  sm += (size_t)256  * 512  * 4;

  auto padwt = [&](const float* W, float* Wt, int Cin, int Cin4, int Cout) {
    const int total = Cout * Cin4;
    pn2_padwt_kernel<<<(total + 255) / 256, 256, 0, stream>>>(W, Wt, Cin, Cin4, Cout);
  };
  auto gemm = [&](const float* X, const float* Wt, float* Y, int R, int Cin, int Cout) {
    dim3 blk(32, 4);
    dim3 grd((R / 16 + 3) / 4, Cout / 64);
    pn2_wmma_gemm_kernel<<<grd, blk, 0, stream>>>(X, Wt, Y, R, Cin, Cout);
  };
  auto bnrelu = [&](float* Y, int R, int C, const float* g, const float* b) {
    pn2_bn_stats_kernel<<<C, 256, 0, stream>>>(Y, R, C, meanb, rinvb);
    const long long total = (long long)R * C;
    pn2_bn_relu_kernel<<<(unsigned)((total + 255) / 256), 256, 0, stream>>>(
        Y, C, meanb, rinvb, g, b, total);
  };

  // Pre-transpose (and K-pad) all GEMM weights.
  padwt(w11, wt11, 6,   8,   64);
  padwt(w12, wt12, 64,  64,  64);
  padwt(w13, wt13, 64,  64,  128);
  padwt(w21, wt21, 131, 132, 128);
  padwt(w22, wt22, 128, 128, 128);
  padwt(w23, wt23, 128, 128, 256);
  padwt(w31, wt31, 259, 260, 256);
  padwt(w32, wt32, 256, 256, 512);
  padwt(w33, wt33, 512, 512, 1024);
  padwt(fc1, wtf1, 1024, 1024, 512);
  padwt(fc2, wtf2, 512, 512, 256);

  // ================= SA1 =================
  pn2_fps_kernel<<<B, N, 0, stream>>>(xyz, fps1, N, M1);
  pn2_gather_xyz_kernel<<<(B * M1 + 255) / 256, 256, 0, stream>>>(xyz, fps1, nx1, N, M1, B * M1);
  pn2_ballq_kernel<<<(B * M1 + 255) / 256, 256, 0, stream>>>(xyz, nx1, gidx1, N, M1, K1, 0.2f * 0.2f, B * M1);
  pn2_group1_kernel<<<(R1 + 255) / 256, 256, 0, stream>>>(xyz, points, nx1, gidx1, P0, N, M1, K1, R1);
  gemm(P0, wt11, P1, R1, 8, 64);    bnrelu(P1, R1, 64, g11, b11);
  gemm(P1, wt12, P0, R1, 64, 64);   bnrelu(P0, R1, 64, g12, b12);
  gemm(P0, wt13, P1, R1, 64, 128);  bnrelu(P1, R1, 128, g13, b13);
  {
    const long long total = (long long)B * M1 * 128;
    pn2_maxpool_kernel<<<(unsigned)((total + 255) / 256), 256, 0, stream>>>(P1, pooled1, K1, 128, total);
  }

  // ================= SA2 =================
  pn2_fps_kernel<<<B, M1, 0, stream>>>(nx1, fps2, M1, M2);
  pn2_gather_xyz_kernel<<<(B * M2 + 255) / 256, 256, 0, stream>>>(nx1, fps2, nx2, M1, M2, B * M2);
  pn2_ballq_kernel<<<(B * M2 + 255) / 256, 256, 0, stream>>>(nx1, nx2, gidx2, M1, M2, K2, 0.4f * 0.4f, B * M2);
  pn2_group2_kernel<<<(R2 + 255) / 256, 256, 0, stream>>>(nx1, pooled1, nx2, gidx2, P0, M1, M2, K2, 128, R2);
  gemm(P0, wt21, P1, R2, 132, 128); bnrelu(P1, R2, 128, g21, b21);
  gemm(P1, wt22, P0, R2, 128, 128); bnrelu(P0, R2, 128, g22, b22);
  gemm(P0, wt23, P1, R2, 128, 256); bnrelu(P1, R2, 256, g23, b23);
  {
    const long long total = (long long)B * M2 * 256;
    pn2_maxpool_kernel<<<(unsigned)((total + 255) / 256), 256, 0, stream>>>(P1, pooled2, K2, 256, total);
  }

  // ================= SA3 (group all) =================
  pn2_group3_kernel<<<(R3 + 255) / 256, 256, 0, stream>>>(nx2, pooled2, P0, 256, R3);
  gemm(P0, wt31, P1, R3, 260, 256);  bnrelu(P1, R3, 256, g31, b31);
  gemm(P1, wt32, P0, R3, 256, 512);  bnrelu(P0, R3, 512, g32, b32);
  gemm(P0, wt33, P1, R3, 512, 1024); bnrelu(P1, R3, 1024, g33, b33);
  {
    const long long total = (long long)B * 1024;   // pool over the 128 points per batch
    pn2_maxpool_kernel<<<(unsigned)((total + 255) / 256), 256, 0, stream>>>(P1, pooled3, 128, 1024, total);
  }

  // ================= Head =================
  gemm(pooled3, wtf1, h1, B, 1024, 512); bnrelu(h1, B, 512, gh1, bh1);
  gemm(h1, wtf2, h2, B, 512, 256);       bnrelu(h2, B, 256, gh2, bh2);
  pn2_cls_kernel<<<(32 * 40 + 255) / 256, 256, 0, stream>>>(h2, clw, clb, out);
}